// UpProjectFastHadamardTransform_80101140070867
// MI455X (gfx1250) — compile-verified
//
#include <hip/hip_runtime.h>

typedef __attribute__((ext_vector_type(2))) float v2f;
typedef __attribute__((ext_vector_type(4))) float v4f;
typedef __attribute__((ext_vector_type(8))) float v8f;
typedef __attribute__((ext_vector_type(4))) int   v4i;

#define D_DIM 8192
#define K_DIM 1024
#define FWHT_SCALE 0.011048543456039806f  /* 1/sqrt(8192) */

// Byte-domain LDS anti-bank-conflict swizzle: XOR bits [7:4] with bits [11:8].
// GF(2)-linear: swb(x^y) == swb(x)^swb(y). Bits [3:0] preserved -> aligned
// 8B/16B groups stay contiguous. All addresses below decompose into
// bit-disjoint fields, so swb distributes over their XOR/sum.
__device__ __forceinline__ constexpr int swb(int a) {
    return a ^ ((a >> 4) & 0xF0);
}

// H16[r][c] = (-1)^popcount(r & c)   (natural-order Walsh-Hadamard)
__device__ __forceinline__ float had16(int r, int c) {
    return (__popc(r & c) & 1) ? -1.0f : 1.0f;
}

__global__ __launch_bounds__(256) void
UpProjectFWHT_kernel(const float* __restrict__ u,
                     const int* __restrict__ idx,
                     float* __restrict__ out) {
    __shared__ __align__(16) float lds[D_DIM];   // 32 KB row buffer (swizzled)
    char* ldsb = (char*)lds;

    const int row  = blockIdx.x;
    const int tid  = threadIdx.x;
    const int lane = tid & 31;
    const int wave = tid >> 5;     // 0..7
    const int n    = lane & 15;    // column within 16x16 tile
    const int h    = lane >> 4;    // lane-half selector

    // ---- 1) zero the row buffer (b128 stores; swizzle is a bijection) -----
    {
        const v4f z = {0.f, 0.f, 0.f, 0.f};
        v4f* l4 = (v4f*)lds;
        #pragma unroll
        for (int i = 0; i < D_DIM / 4 / 256; ++i) l4[tid + i * 256] = z;
    }
    __syncthreads();

    // ---- 2) scatter-add u[row,:] at idx (dups -> ds_add_f32 atomics) ------
    {
        const v4i i4 = ((const v4i*)idx)[tid];                     // 4 indices
        const v4f u4 = ((const v4f*)(u + (size_t)row * K_DIM))[tid];
        atomicAdd((float*)(ldsb + swb(4 * i4.x)), u4.x);
        atomicAdd((float*)(ldsb + swb(4 * i4.y)), u4.y);
        atomicAdd((float*)(ldsb + swb(4 * i4.z)), u4.z);
        atomicAdd((float*)(ldsb + swb(4 * i4.w)), u4.w);
    }
    __syncthreads();

    // ---- 3) three radix-16 stages: D = H16 x M via V_WMMA_F32_16X16X4_F32 -
    // element index = b*4096 + i*256 + j*16 + k; stages transform k, j, i.
    // A (16x4 f32): lane row n, K-cols 2h,2h+1 per chunk. Pin the 8 signs in
    // VGPRs with empty asm so they aren't rematerialized at every WMMA.
    float sgn[8];
    #pragma unroll
    for (int ck = 0; ck < 4; ++ck) {
        sgn[2 * ck]     = had16(n, 4 * ck + 2 * h);
        sgn[2 * ck + 1] = had16(n, 4 * ck + 2 * h + 1);
    }
    #pragma unroll
    for (int i = 0; i < 8; ++i) asm volatile("" : "+v"(sgn[i]));
    v2f aH[4];
    #pragma unroll
    for (int ck = 0; ck < 4; ++ck) {
        aH[ck].x = sgn[2 * ck];
        aH[ck].y = sgn[2 * ck + 1];
    }

    #pragma unroll
    for (int stage = 0; stage < 3; ++stage) {
        const int RS = (stage == 0) ? 4 : (stage == 1) ? 64 : 1024; // row bytes
        const int CS = (stage == 0) ? 64 : 4;                       // col bytes
        // Per-lane swizzled components (computed once per stage):
        const int laneLoad  = swb(CS * n) ^ swb(RS * 2 * h); // B rows 2h,2h+1
        const int laneStore = swb(CS * n) ^ swb(RS * 8 * h); // D rows 8h..8h+7

        #pragma unroll
        for (int q = 0; q < 4; ++q) {
            const int t = wave * 4 + q;   // tile 0..31, disjoint per wave
            const int baseB = (stage < 2)
                ? ((t >> 4) * 16384 + (t & 15) * 1024)   // (b, i) tile
                : ((t >> 4) * 16384 + (t & 15) * 64);    // (b, j) tile
            const int sB = swb(baseB);                   // wave-uniform (SALU)
            const int lb = sB ^ laneLoad;

            v8f acc = {};
            #pragma unroll
            for (int ck = 0; ck < 4; ++ck) {
                v2f b;
                if (stage == 0) {
                    // rows contiguous: one 8B LDS load; row term = compile const
                    b = *(const v2f*)(ldsb + (lb ^ swb(16 * ck)));
                } else {
                    b.x = *(const float*)(ldsb + (lb ^ swb(4 * RS * ck)));
                    b.y = *(const float*)(ldsb + (lb ^ swb(4 * RS * ck) ^ swb(RS)));
                }
                // (neg_a, A, neg_b, B, c_mod, C, reuse_a, reuse_b)
                acc = __builtin_amdgcn_wmma_f32_16x16x4_f32(
                        false, aH[ck], false, b, (short)0, acc, false, false);
            }

            const int sb2 = sB ^ laneStore;
            if (stage == 0) {
                v4f s0 = {acc[0], acc[1], acc[2], acc[3]};
                v4f s1 = {acc[4], acc[5], acc[6], acc[7]};
                *(v4f*)(ldsb + sb2)              = s0;   // rows 8h..8h+3
                *(v4f*)(ldsb + (sb2 ^ swb(16)))  = s1;   // rows 8h+4..8h+7
            } else {
                #pragma unroll
                for (int v = 0; v < 8; ++v)
                    *(float*)(ldsb + (sb2 ^ swb(RS * v))) = acc[v];
            }
        }
        __syncthreads();
    }

    // ---- 4) radix-2 (stride 4096 floats) fused with scale + NT b128 stores -
    float* orow = out + (size_t)row * D_DIM;
    const int laneF = swb(16 * tid);          // per-lane component, once
    #pragma unroll
    for (int s = 0; s < 4; ++s) {
        const int ax = laneF ^ swb(4096 * s);            // bytes, disjoint fields
        const v4f x = *(const v4f*)(ldsb + ax);
        const v4f y = *(const v4f*)(ldsb + (ax ^ swb(16384)));
        v4f o0 = {(x.x + y.x) * FWHT_SCALE, (x.y + y.y) * FWHT_SCALE,
                  (x.z + y.z) * FWHT_SCALE, (x.w + y.w) * FWHT_SCALE};
        v4f o1 = {(x.x - y.x) * FWHT_SCALE, (x.y - y.y) * FWHT_SCALE,
                  (x.z - y.z) * FWHT_SCALE, (x.w - y.w) * FWHT_SCALE};
        const int m = s * 1024 + tid * 4;                // output float index
        __builtin_nontemporal_store(o0, (v4f*)(orow + m));
        __builtin_nontemporal_store(o1, (v4f*)(orow + m + 4096));
    }
}

extern "C" void kernel_launch(void* const* d_in, const int* in_sizes, int n_in,
                              void* d_out, int out_size, void* d_ws, size_t ws_size,
                              hipStream_t stream) {
    (void)n_in; (void)out_size; (void)d_ws; (void)ws_size;
    const float* u   = (const float*)d_in[0];
    const int*   idx = (const int*)d_in[1];
    float*       out = (float*)d_out;

    const int rows = in_sizes[0] / K_DIM;   // 4*2048 = 8192
    UpProjectFWHT_kernel<<<rows, 256, 0, stream>>>(u, idx, out);
}